// Decoding_model_10230612099661
// MI455X (gfx1250) — compile-verified
//
#include <hip/hip_runtime.h>
#include <hip/hip_bf16.h>

// LDPC normalized-min-sum BP decoder for MI455X (gfx1250).
// B=128 blocks, one per codeword; soft state lives in LDS; TDM moves rows.

typedef unsigned int v4ui __attribute__((ext_vector_type(4)));
typedef int          v4si __attribute__((ext_vector_type(4)));
typedef int          v8si __attribute__((ext_vector_type(8)));

#define LARGE_F 1e10f
#define N_VARS  1024
#define M_CHK   512
#define ROW_W   8

// ---- TDM descriptor builders (CDNA5 ISA) ----
// Group0: [1:0]=count=1, [63:32]=lds_addr, [120:64]=global_addr, [127:126]=type=2
__device__ __forceinline__ v4ui tdm_g0(unsigned lds_addr, const void* gp) {
  unsigned long long ga = (unsigned long long)gp;
  v4ui g0;
  g0.x = 1u;                                   // count=1 (valid), no gather
  g0.y = lds_addr;                             // LDS byte address
  g0.z = (unsigned)(ga & 0xFFFFFFFFull);       // global_addr[31:0]
  g0.w = (unsigned)((ga >> 32) & 0x01FFFFFFull) | (2u << 30); // addr[56:32] | type=2
  return g0;
}

// Group1: 1-D tile: data_size=4B, tensor_dim0=n, tensor_dim1=1,
// tile_dim0=n, tile_dim1=1, tensor_dim0_stride=n
__device__ __forceinline__ v8si tdm_g1(unsigned n) {
  v8si g1;
  g1[0] = 0x00020000;                              // wg_mask=0 | data_size=2 (4B)
  g1[1] = (int)((n & 0xFFFFu) << 16);              // tensor_dim0[15:0]
  g1[2] = (int)(((n >> 16) & 0xFFFFu) | (1u<<16)); // td0[31:16] | tensor_dim1[15:0]=1
  g1[3] = (int)((n & 0xFFFFu) << 16);              // td1[31:16]=0 | tile_dim0=n
  g1[4] = 1;                                       // tile_dim1=1 | tile_dim2=0
  g1[5] = (int)n;                                  // tensor_dim0_stride[31:0]
  g1[6] = (int)((n & 0xFFFFu) << 16);              // stride0[47:32]=0 | stride1[15:0]=n
  g1[7] = 0;                                       // stride1[47:16]=0
  return g1;
}

__device__ __forceinline__ void tdm_load_row(unsigned lds_addr, const float* gsrc,
                                             unsigned n) {
  v4ui g0 = tdm_g0(lds_addr, (const void*)gsrc);
  v8si g1 = tdm_g1(n);
  v4si gz = {0, 0, 0, 0};
#if __clang_major__ >= 23
  v8si gz8 = {0, 0, 0, 0, 0, 0, 0, 0};
  __builtin_amdgcn_tensor_load_to_lds(g0, g1, gz, gz, gz8, 0);
#else
  __builtin_amdgcn_tensor_load_to_lds(g0, g1, gz, gz, 0);
#endif
}

__device__ __forceinline__ void tdm_store_row(unsigned lds_addr, float* gdst,
                                              unsigned n) {
  v4ui g0 = tdm_g0(lds_addr, (const void*)gdst);
  v8si g1 = tdm_g1(n);
  v4si gz = {0, 0, 0, 0};
#if __clang_major__ >= 23
  v8si gz8 = {0, 0, 0, 0, 0, 0, 0, 0};
  __builtin_amdgcn_tensor_store_from_lds(g0, g1, gz, gz, gz8, 0);
#else
  __builtin_amdgcn_tensor_store_from_lds(g0, g1, gz, gz, 0);
#endif
}

// ---- preprocess: dense H [m,n] -> per-check column lists [m, ROW_W] ----
__global__ void extract_cols_kernel(const int* __restrict__ H,
                                    int* __restrict__ cols,
                                    int m, int n) {
  int r = blockIdx.x * blockDim.x + threadIdx.x;
  if (r >= m) return;
  const int* row = H + (size_t)r * n;
  int k = 0;
  for (int c = 0; c < n; ++c) {
    if (row[c] != 0 && k < ROW_W) cols[r * ROW_W + (k++)] = c;
  }
  for (; k < ROW_W; ++k) cols[r * ROW_W + k] = -1;   // sentinel (never hit here)
}

// ---- main BP kernel: one workgroup (8 wave32) per batch element ----
__global__ __launch_bounds__(256)
void bp_main_kernel(const float* __restrict__ soft_in,
                    const float* __restrict__ normalizer,
                    const int*   __restrict__ niter,
                    const int*   __restrict__ cols_g,
                    float* __restrict__ out,
                    int B, int m, int n) {
  __shared__ float s_soft[N_VARS];
  __shared__ float s_delta[N_VARS];
  __shared__ int   s_cols[M_CHK * ROW_W];

  const int tid = threadIdx.x;
  const int b   = blockIdx.x;
  const int T   = niter[0];
  const float norm = log1pf(expf(normalizer[0]));   // softplus(normalizer)

  // cache check adjacency in LDS
  for (int i = tid; i < m * ROW_W; i += blockDim.x) s_cols[i] = cols_g[i];

  // LDS byte address: generic-pointer low 32 bits are the LDS offset (aperture in high bits)
  const unsigned lds_soft = (unsigned)(unsigned long long)(void*)&s_soft[0];

  // async tensor DMA: soft_input row -> LDS
  if (tid < 32) {
    tdm_load_row(lds_soft, soft_in + (size_t)b * n, (unsigned)n);
    __builtin_amdgcn_s_wait_tensorcnt(0);
  }
  __syncthreads();

  // snapshot 0: soft_input (LDS -> out row 0, async)
  if (tid < 32) tdm_store_row(lds_soft, out + (size_t)b * n, (unsigned)n);

  for (int it = 1; it <= T; ++it) {
    for (int i = tid; i < n; i += blockDim.x) s_delta[i] = 0.0f;
    __syncthreads();

    // 512 checks / 256 threads = 2 checks each
    for (int c = tid; c < m; c += blockDim.x) {
      const int* cp = &s_cols[c * ROW_W];
      float v[ROW_W];
      int   cl[ROW_W];
      unsigned parity = 0;
      bool anyzero = false;
      float min1 = LARGE_F, min2 = LARGE_F;
#pragma unroll
      for (int e = 0; e < ROW_W; ++e) {
        int cc = cp[e];
        cl[e] = cc;
        float vv = (cc >= 0) ? s_soft[cc] : 0.0f;
        v[e] = vv;
        if (cc >= 0) {
          if (vv == 0.0f) anyzero = true;      // jnp.sign(0)=0 zeroes whole row
          parity ^= (__float_as_uint(vv) >> 31);
          float a  = fabsf(vv);
          float am = (vv == 0.0f) ? LARGE_F : a;  // proc: zero-mag -> LARGE
          if (am < min1) { min2 = min1; min1 = am; }
          else if (am < min2) { min2 = am; }
        }
      }
      if (!anyzero) {
#pragma unroll
        for (int e = 0; e < ROW_W; ++e) {
          int cc = cl[e];
          if (cc < 0) continue;
          float a   = fabsf(v[e]);
          float mag = (a == min1) ? min2 : min1;   // avc==smallest -> second
          unsigned sb = parity ^ (__float_as_uint(v[e]) >> 31); // extrinsic sign
          float cv = norm * mag;
          cv = sb ? -cv : cv;
          atomicAdd(&s_delta[cc], cv);             // ds_add_f32
        }
      }
    }
    __syncthreads();

    // previous async store must finish reading s_soft before we mutate it
    if (tid < 32) __builtin_amdgcn_s_wait_tensorcnt(0);
    __syncthreads();

    for (int i = tid; i < n; i += blockDim.x) s_soft[i] += s_delta[i];
    __syncthreads();

    // snapshot `it`: LDS -> out[it*B*n + b*n], async (overlaps next iteration)
    if (tid < 32)
      tdm_store_row(lds_soft, out + ((size_t)it * B + b) * n, (unsigned)n);
  }

  if (tid < 32) __builtin_amdgcn_s_wait_tensorcnt(0);
}

// ---- labels pass-through (int -> float into tail of d_out) ----
__global__ void copy_labels_kernel(const int* __restrict__ lab,
                                   float* __restrict__ dst, int count) {
  int i = blockIdx.x * blockDim.x + threadIdx.x;
  if (i < count) dst[i] = (float)lab[i];
}

extern "C" void kernel_launch(void* const* d_in, const int* in_sizes, int n_in,
                              void* d_out, int out_size, void* d_ws, size_t ws_size,
                              hipStream_t stream) {
  const float* soft       = (const float*)d_in[0];
  const int*   labels     = (const int*)d_in[1];
  const int*   H          = (const int*)d_in[2];
  const float* normalizer = (const float*)d_in[3];
  const int*   niter      = (const int*)d_in[4];

  const int n = N_VARS;
  const int B = in_sizes[0] / n;   // 128
  const int m = in_sizes[2] / n;   // 512
  int* cols = (int*)d_ws;          // m*ROW_W ints = 16 KB scratch

  extract_cols_kernel<<<(m + 255) / 256, 256, 0, stream>>>(H, cols, m, n);
  bp_main_kernel<<<B, 256, 0, stream>>>(soft, normalizer, niter, cols,
                                        (float*)d_out, B, m, n);

  const int lab_count = in_sizes[1];
  float* lab_dst = (float*)d_out + ((size_t)out_size - (size_t)lab_count);
  copy_labels_kernel<<<(lab_count + 255) / 256, 256, 0, stream>>>(labels, lab_dst,
                                                                  lab_count);
}